// AnchorGNN_62715112456878
// MI455X (gfx1250) — compile-verified
//
#include <hip/hip_runtime.h>
#include <math.h>

typedef float v2f __attribute__((ext_vector_type(2)));
typedef float v8f __attribute__((ext_vector_type(8)));

#define DEV static __device__ __forceinline__

namespace {
constexpr int N_NODES = 50000;
constexpr int NT      = N_NODES / 16;   // 3125 tiles, exact
constexpr int NFEAT   = 256;
constexpr int NHID    = 128;
constexpr int NA      = 32;
constexpr int NP      = 4;
constexpr float EPSV   = 1e-12f;
constexpr float LN_EPS = 1e-5f;

// workspace layout (floats)
constexpr size_t WS_H    = 0;                                  // N x 128
constexpr size_t WS_XC   = WS_H    + (size_t)N_NODES * NHID;   // N x 128
constexpr size_t WS_ADJ  = WS_XC   + (size_t)N_NODES * NHID;   // N x 32
constexpr size_t WS_INVR = WS_ADJ  + (size_t)N_NODES * NA;     // N
constexpr size_t WS_CS   = WS_INVR + (size_t)N_NODES;          // 32
constexpr size_t WS_HAB  = WS_CS   + 32;                       // P*A*128
constexpr size_t WS_S0   = WS_HAB  + (size_t)NP * NA * NHID;   // 32x128
constexpr size_t WS_S1   = WS_S0   + (size_t)NA * NHID;        // 32x128
constexpr size_t WS_AGG  = WS_S1   + (size_t)NA * NHID;        // 32x128
} // namespace

DEV void wait_ds() { asm volatile("s_wait_dscnt 0" ::: "memory"); }
DEV void wait_async() { asm volatile("s_wait_asynccnt 0" ::: "memory"); }

// Async DMA 16B/lane global -> LDS (GV mode), tracked by ASYNCcnt.
// Generic LDS pointers carry the wave-relative LDS byte offset in addr[31:0].
DEV void async_ld_b128(float* lds_ptr, const float* gptr) {
  const unsigned lds_off = (unsigned)(size_t)(void*)lds_ptr;
  asm volatile("global_load_async_to_lds_b128 %0, %1, off"
               :: "v"(lds_off), "v"((unsigned long long)(size_t)gptr)
               : "memory");
}

// D(16x16,f32) = A(16x4,f32) * B(4x16,f32) + C   -- native fp32 WMMA
DEV v8f wmma4(v2f a, v2f b, v8f c) {
  return __builtin_amdgcn_wmma_f32_16x16x4_f32(false, a, false, b, (short)0, c,
                                               false, false);
}

DEV float xor_reduce32(float v) {
  v += __shfl_xor(v, 16, 32);
  v += __shfl_xor(v, 8, 32);
  v += __shfl_xor(v, 4, 32);
  v += __shfl_xor(v, 2, 32);
  v += __shfl_xor(v, 1, 32);
  return v;
}

// ---------------------------------------------------------------------------
// k_prep: haB[p,a,:] = anchors@Wa[p] + ml_b1[p]; anchor classifier outputs;
// zero colsum / S0 / S1 (must run every launch: graph replay determinism).
// ---------------------------------------------------------------------------
__global__ void __launch_bounds__(128) k_prep(
    const float* __restrict__ anchors, const float* __restrict__ mlw1,
    const float* __restrict__ mlb1, const float* __restrict__ cw1,
    const float* __restrict__ cb1, const float* __restrict__ pa,
    const float* __restrict__ cw2, const float* __restrict__ cb2,
    float* __restrict__ haB, float* __restrict__ colsum, float* __restrict__ S0,
    float* __restrict__ S1, float* __restrict__ avOut, float* __restrict__ alOut) {
  __shared__ float t1[NA * NHID];
  const int t = threadIdx.x;
  for (int i = t; i < NA; i += 128) colsum[i] = 0.f;
  for (int i = t; i < NA * NHID; i += 128) { S0[i] = 0.f; S1[i] = 0.f; }
  for (int i = t; i < NA * NHID; i += 128) avOut[i] = anchors[i];
  // haB
  for (int i = t; i < NP * NA * NHID; i += 128) {
    const int p = i >> 12, rem = i & 4095, a = rem >> 7, j = rem & 127;
    float s = mlb1[p * NHID + j];
    for (int k = 0; k < NHID; ++k)
      s += anchors[a * NHID + k] * mlw1[((size_t)p * 2 * NHID + NHID + k) * NHID + j];
    haB[i] = s;
  }
  // anchor classifier hidden
  for (int i = t; i < NA * NHID; i += 128) {
    const int a = i >> 7, j = i & 127;
    float s = cb1[j];
    for (int k = 0; k < NHID; ++k) s += anchors[a * NHID + k] * cw1[k * NHID + j];
    t1[i] = s > 0.f ? s : pa[j] * s;
  }
  __syncthreads();
  for (int i = t; i < NA * 2; i += 128) {
    const int a = i >> 1, c = i & 1;
    float s = cb2[c];
    for (int k = 0; k < NHID; ++k) s += t1[a * NHID + k] * cw2[k * 2 + c];
    alOut[i] = s;
  }
}

// ---------------------------------------------------------------------------
// k_embed: per 16-node tile (one wave each):
//   h = relu(X@W1+b1)@W2+b2  (WMMA f32), attention adj row, rowsum, colsum.
//   X tile and ml_w2 staged into LDS with async global->LDS DMA.
// ---------------------------------------------------------------------------
__global__ void __launch_bounds__(64) k_embed(
    const float* __restrict__ x, const float* __restrict__ w1,
    const float* __restrict__ b1, const float* __restrict__ w2,
    const float* __restrict__ b2, const float* __restrict__ haB,
    const float* __restrict__ mlw1, const float* __restrict__ mlw2,
    const float* __restrict__ mlb2, float* __restrict__ hG,
    float* __restrict__ adjG, float* __restrict__ invrsG,
    float* __restrict__ colsum) {
  constexpr int LDX = NFEAT + 4;  // 260 (bank-conflict-free)
  constexpr int LDT = NHID + 4;   // 132
  constexpr int PW = 16 * LDX + 16 * LDT + NP * NHID;  // floats per wave
  __shared__ __align__(16) float lds[2 * PW];
  const int wave = threadIdx.x >> 5, lane = threadIdx.x & 31;
  const int r = lane & 15, half = lane >> 4;
  float* xst = lds + wave * PW;       // 16 x 260 : X tile, later h tile
  float* tb  = xst + 16 * LDX;        // 16 x 132 : t tile, later hx tile
  float* w2s = tb + 16 * LDT;         // P x 128  : ml_w2 staged
  const int tile = blockIdx.x * 2 + wave;
  if (tile >= NT) return;
  const int n0 = tile * 16;

  // async-DMA stage: ml_w2 (P*128 floats) and X tile (16x256) into LDS
  for (int i = lane; i < NP * NHID / 4; i += 32)
    async_ld_b128(w2s + i * 4, mlw2 + i * 4);
  for (int i = lane; i < 16 * (NFEAT / 4); i += 32) {
    const int row = i >> 6, c = (i & 63) << 2;
    async_ld_b128(xst + row * LDX + c, x + (size_t)(n0 + row) * NFEAT + c);
  }
  // prefetch haB (reused 16x per tile later in the attention loop)
  for (int i = lane; i < NP * NA; i += 32)
    __builtin_prefetch(haB + (size_t)i * NHID, 0, 0);
  wait_async();

  // t = relu(X @ W1 + b1) -> tb
  for (int nt = 0; nt < 8; ++nt) {
    v8f acc = {0, 0, 0, 0, 0, 0, 0, 0};
#pragma unroll 8
    for (int k0 = 0; k0 < NFEAT; k0 += 4) {
      const int kk = k0 + 2 * half;
      v2f a = *(const v2f*)(xst + r * LDX + kk);
      v2f b;
      b.x = w1[(size_t)kk * NHID + nt * 16 + r];
      b.y = w1[(size_t)(kk + 1) * NHID + nt * 16 + r];
      acc = wmma4(a, b, acc);
    }
    const float bias = b1[nt * 16 + r];
#pragma unroll
    for (int q = 0; q < 8; ++q)
      tb[(q + 8 * half) * LDT + nt * 16 + r] = fmaxf(acc[q] + bias, 0.f);
  }
  wait_ds();

  // h = t @ W2 + b2 -> xst (reuse X slab), no relu
  for (int nt = 0; nt < 8; ++nt) {
    v8f acc = {0, 0, 0, 0, 0, 0, 0, 0};
#pragma unroll 8
    for (int k0 = 0; k0 < NHID; k0 += 4) {
      const int kk = k0 + 2 * half;
      v2f a = *(const v2f*)(tb + r * LDT + kk);
      v2f b;
      b.x = w2[(size_t)kk * NHID + nt * 16 + r];
      b.y = w2[(size_t)(kk + 1) * NHID + nt * 16 + r];
      acc = wmma4(a, b, acc);
    }
    const float bias = b2[nt * 16 + r];
#pragma unroll
    for (int q = 0; q < 8; ++q)
      xst[(q + 8 * half) * LDX + nt * 16 + r] = acc[q] + bias;
  }
  wait_ds();

  // write h tile to global (coalesced from LDS)
  for (int i = lane; i < 16 * (NHID / 4); i += 32) {
    const int row = i >> 5, c = (i & 31) << 2;
    *(float4*)(hG + (size_t)(n0 + row) * NHID + c) =
        *(const float4*)(xst + row * LDX + c);
  }

  // attention: adj[n,a] = (1/P) * sum_p sigmoid(relu(hx + haB)·w2 + b2)
  float adjacc[16];
#pragma unroll
  for (int i = 0; i < 16; ++i) adjacc[i] = 0.f;

  for (int p = 0; p < NP; ++p) {
    const float* Wx = mlw1 + (size_t)p * 2 * NHID * NHID;  // rows 0..127
    for (int nt = 0; nt < 8; ++nt) {
      v8f acc = {0, 0, 0, 0, 0, 0, 0, 0};
#pragma unroll 8
      for (int k0 = 0; k0 < NHID; k0 += 4) {
        const int kk = k0 + 2 * half;
        v2f a = *(const v2f*)(xst + r * LDX + kk);  // h
        v2f b;
        b.x = Wx[(size_t)kk * NHID + nt * 16 + r];
        b.y = Wx[(size_t)(kk + 1) * NHID + nt * 16 + r];
        acc = wmma4(a, b, acc);
      }
#pragma unroll
      for (int q = 0; q < 8; ++q)
        tb[(q + 8 * half) * LDT + nt * 16 + r] = acc[q];  // hx
    }
    wait_ds();
    const float* hab = haB + (size_t)p * NA * NHID;
    const float mb2 = mlb2[p];
#pragma unroll
    for (int node = 0; node < 16; ++node) {
      const float hx0 = tb[node * LDT + lane];
      const float hx1 = tb[node * LDT + 32 + lane];
      const float hx2 = tb[node * LDT + 64 + lane];
      const float hx3 = tb[node * LDT + 96 + lane];
      const float w20 = w2s[p * NHID + lane];
      const float w21 = w2s[p * NHID + 32 + lane];
      const float w22 = w2s[p * NHID + 64 + lane];
      const float w23 = w2s[p * NHID + 96 + lane];
      for (int a = 0; a < NA; ++a) {
        const float* hb = hab + (size_t)a * NHID;
        float s = fmaxf(hx0 + hb[lane], 0.f) * w20 +
                  fmaxf(hx1 + hb[32 + lane], 0.f) * w21 +
                  fmaxf(hx2 + hb[64 + lane], 0.f) * w22 +
                  fmaxf(hx3 + hb[96 + lane], 0.f) * w23;
        s = xor_reduce32(s);
        const float sig = 1.f / (1.f + __expf(-(s + mb2)));
        if (lane == a) adjacc[node] += sig;  // lane a owns anchor a
      }
    }
  }

  // write adj (lane=anchor), colsum atomics, inverse rowsum
  float csum = 0.f;
#pragma unroll
  for (int node = 0; node < 16; ++node) {
    const float adjv = adjacc[node] * (1.f / NP);
    adjG[(size_t)(n0 + node) * NA + lane] = adjv;
    csum += adjv;
    const float rs = xor_reduce32(adjv);
    if (lane == node) invrsG[n0 + node] = 1.f / fmaxf(rs, EPSV);
  }
  atomicAdd(&colsum[lane], csum);
}

// ---------------------------------------------------------------------------
// k_hop_gather: support = xc @ enc_w (WMMA); S += adj^T @ support (WMMA,
// persistent 2x8 v8f accumulators per wave), atomic flush at end.
// xc tile staged with async global->LDS DMA; next tile prefetched.
// ---------------------------------------------------------------------------
__global__ void __launch_bounds__(64) k_hop_gather(
    const float* __restrict__ xc, const float* __restrict__ adjG,
    const float* __restrict__ encw, float* __restrict__ S) {
  constexpr int LDT = NHID + 4;
  constexpr int PW = 2 * 16 * LDT;
  __shared__ __align__(16) float lds[2 * PW];
  const int wave = threadIdx.x >> 5, lane = threadIdx.x & 31;
  const int r = lane & 15, half = lane >> 4;
  float* xst = lds + wave * PW;   // 16x132 xc tile
  float* sup = xst + 16 * LDT;    // 16x132 support tile
  v8f s[16];
#pragma unroll
  for (int i = 0; i < 16; ++i) s[i] = (v8f){0, 0, 0, 0, 0, 0, 0, 0};

  const int nw = gridDim.x * 2;
  for (int tile = blockIdx.x * 2 + wave; tile < NT; tile += nw) {
    const int n0 = tile * 16;
    wait_ds();  // WAR: previous iteration's LDS reads before async overwrite
    for (int i = lane; i < 16 * (NHID / 4); i += 32) {
      const int row = i >> 5, c = (i & 31) << 2;
      async_ld_b128(xst + row * LDT + c, xc + (size_t)(n0 + row) * NHID + c);
    }
    // speculative prefetch of next tile's xc rows and adj rows
    if (tile + nw < NT) {
      const size_t nn0 = (size_t)(tile + nw) * 16;
      __builtin_prefetch(xc + (nn0 + (lane >> 1)) * NHID + (lane & 1) * 64, 0, 0);
      __builtin_prefetch(adjG + nn0 * NA + lane * 16, 0, 0);
    }
    wait_async();
    for (int nt = 0; nt < 8; ++nt) {
      v8f acc = {0, 0, 0, 0, 0, 0, 0, 0};
#pragma unroll 8
      for (int k0 = 0; k0 < NHID; k0 += 4) {
        const int kk = k0 + 2 * half;
        v2f a = *(const v2f*)(xst + r * LDT + kk);
        v2f b;
        b.x = encw[(size_t)kk * NHID + nt * 16 + r];
        b.y = encw[(size_t)(kk + 1) * NHID + nt * 16 + r];
        acc = wmma4(a, b, acc);
      }
#pragma unroll
      for (int q = 0; q < 8; ++q)
        sup[(q + 8 * half) * LDT + nt * 16 + r] = acc[q];
    }
    wait_ds();
    // S(32x128) += adj^T(32x16) @ support(16x128)
#pragma unroll
    for (int mt = 0; mt < 2; ++mt) {
#pragma unroll
      for (int ntt = 0; ntt < 8; ++ntt) {
        v8f acc = s[mt * 8 + ntt];
#pragma unroll
        for (int k0 = 0; k0 < 16; k0 += 4) {
          const int kk = k0 + 2 * half;
          v2f a;  // A[m=anchor][k=node] = adj[node][anchor]
          a.x = adjG[(size_t)(n0 + kk) * NA + mt * 16 + r];
          a.y = adjG[(size_t)(n0 + kk + 1) * NA + mt * 16 + r];
          v2f b;  // B[k=node][n=j] = support[node][j]
          b.x = sup[kk * LDT + ntt * 16 + r];
          b.y = sup[(kk + 1) * LDT + ntt * 16 + r];
          acc = wmma4(a, b, acc);
        }
        s[mt * 8 + ntt] = acc;
      }
    }
  }
#pragma unroll
  for (int mt = 0; mt < 2; ++mt)
#pragma unroll
    for (int ntt = 0; ntt < 8; ++ntt)
#pragma unroll
      for (int q = 0; q < 8; ++q)
        atomicAdd(&S[(size_t)(mt * 16 + q + 8 * half) * NHID + ntt * 16 + r],
                  s[mt * 8 + ntt][q]);
}

// agg = S / max(colsum, eps)
__global__ void __launch_bounds__(128) k_hop_mid(const float* __restrict__ S,
                                                 const float* __restrict__ colsum,
                                                 float* __restrict__ agg) {
  const int i = blockIdx.x * blockDim.x + threadIdx.x;
  if (i < NA * NHID) agg[i] = S[i] / fmaxf(colsum[i >> 7], EPSV);
}

// ---------------------------------------------------------------------------
// k_hop_apply: xc' = relu((adj*invrowsum) @ agg + enc_b)  (WMMA).
// FINAL: z = xc'+h, LayerNorm, classifier (WMMA) -> pred & z outputs.
// ---------------------------------------------------------------------------
template <bool FINAL>
__global__ void __launch_bounds__(64) k_hop_apply(
    const float* __restrict__ adjG, const float* __restrict__ invrsG,
    const float* __restrict__ agg, const float* __restrict__ encb,
    const float* __restrict__ hG, const float* __restrict__ lng,
    const float* __restrict__ lnb, const float* __restrict__ cw1,
    const float* __restrict__ cb1, const float* __restrict__ pa,
    const float* __restrict__ cw2, const float* __restrict__ cb2,
    float* __restrict__ xcOut, float* __restrict__ predOut,
    float* __restrict__ zOut) {
  constexpr int LDT = NHID + 4;
  constexpr int PW = 2 * 16 * LDT;
  __shared__ __align__(16) float lds[2 * PW];
  const int wave = threadIdx.x >> 5, lane = threadIdx.x & 31;
  const int r = lane & 15, half = lane >> 4;
  float* zb = lds + wave * PW;
  float* tb = zb + 16 * LDT;
  const int tile = blockIdx.x * 2 + wave;
  if (tile >= NT) return;
  const int n0 = tile * 16;
  const float invr = invrsG[n0 + r];

  for (int nt = 0; nt < 8; ++nt) {
    v8f acc = {0, 0, 0, 0, 0, 0, 0, 0};
#pragma unroll
    for (int k0 = 0; k0 < NA; k0 += 4) {
      const int kk = k0 + 2 * half;
      v2f a = *(const v2f*)(adjG + (size_t)(n0 + r) * NA + kk);
      a.x *= invr;
      a.y *= invr;
      v2f b;
      b.x = agg[kk * NHID + nt * 16 + r];
      b.y = agg[(kk + 1) * NHID + nt * 16 + r];
      acc = wmma4(a, b, acc);
    }
    const float bias = encb[nt * 16 + r];
#pragma unroll
    for (int q = 0; q < 8; ++q) {
      float v = fmaxf(acc[q] + bias, 0.f);
      if (FINAL) {
        v += hG[(size_t)(n0 + q + 8 * half) * NHID + nt * 16 + r];  // residual
        zb[(q + 8 * half) * LDT + nt * 16 + r] = v;
      } else {
        xcOut[(size_t)(n0 + q + 8 * half) * NHID + nt * 16 + r] = v;
      }
    }
  }
  if (!FINAL) return;
  wait_ds();

  // LayerNorm: lane -> node=r, half of features=half
  {
    float sum = 0.f;
    for (int j = half * 64; j < half * 64 + 64; ++j) sum += zb[r * LDT + j];
    sum += __shfl_xor(sum, 16, 32);
    const float mu = sum * (1.f / NHID);
    float vs = 0.f;
    for (int j = half * 64; j < half * 64 + 64; ++j) {
      const float d = zb[r * LDT + j] - mu;
      vs += d * d;
    }
    vs += __shfl_xor(vs, 16, 32);
    const float rstd = rsqrtf(vs * (1.f / NHID) + LN_EPS);
    for (int j = half * 64; j < half * 64 + 64; ++j)
      zb[r * LDT + j] = (zb[r * LDT + j] - mu) * rstd * lng[j] + lnb[j];
  }
  wait_ds();

  // write normalized z (coalesced)
  for (int i = lane; i < 16 * (NHID / 4); i += 32) {
    const int row = i >> 5, c = (i & 31) << 2;
    *(float4*)(zOut + (size_t)(n0 + row) * NHID + c) =
        *(const float4*)(zb + row * LDT + c);
  }

  // t1 = prelu(z @ cls_w1 + cls_b1) -> tb
  for (int nt = 0; nt < 8; ++nt) {
    v8f acc = {0, 0, 0, 0, 0, 0, 0, 0};
#pragma unroll 8
    for (int k0 = 0; k0 < NHID; k0 += 4) {
      const int kk = k0 + 2 * half;
      v2f a = *(const v2f*)(zb + r * LDT + kk);
      v2f b;
      b.x = cw1[(size_t)kk * NHID + nt * 16 + r];
      b.y = cw1[(size_t)(kk + 1) * NHID + nt * 16 + r];
      acc = wmma4(a, b, acc);
    }
    const float bias = cb1[nt * 16 + r], aj = pa[nt * 16 + r];
#pragma unroll
    for (int q = 0; q < 8; ++q) {
      float v = acc[q] + bias;
      v = v > 0.f ? v : aj * v;
      tb[(q + 8 * half) * LDT + nt * 16 + r] = v;
    }
  }
  wait_ds();

  // pred = t1 @ cls_w2 + cls_b2 : lane -> (node=r, class=half)
  {
    float acc2 = cb2[half];
    for (int j = 0; j < NHID; ++j) acc2 += tb[r * LDT + j] * cw2[j * 2 + half];
    predOut[(size_t)(n0 + r) * 2 + half] = acc2;
  }
}

// ---------------------------------------------------------------------------
extern "C" void kernel_launch(void* const* d_in, const int* in_sizes, int n_in,
                              void* d_out, int out_size, void* d_ws,
                              size_t ws_size, hipStream_t stream) {
  (void)in_sizes; (void)n_in; (void)out_size; (void)ws_size;
  const float* x       = (const float*)d_in[0];
  const float* im_w1   = (const float*)d_in[1];
  const float* im_b1   = (const float*)d_in[2];
  const float* im_w2   = (const float*)d_in[3];
  const float* im_b2   = (const float*)d_in[4];
  const float* anchors = (const float*)d_in[5];
  const float* ml_w1   = (const float*)d_in[6];
  const float* ml_b1   = (const float*)d_in[7];
  const float* ml_w2   = (const float*)d_in[8];
  const float* ml_b2   = (const float*)d_in[9];
  const float* enc_w   = (const float*)d_in[10];
  const float* enc_b   = (const float*)d_in[11];
  const float* ln_g    = (const float*)d_in[12];
  const float* ln_b    = (const float*)d_in[13];
  const float* cls_w1  = (const float*)d_in[14];
  const float* cls_b1  = (const float*)d_in[15];
  const float* prelu_a = (const float*)d_in[16];
  const float* cls_w2  = (const float*)d_in[17];
  const float* cls_b2  = (const float*)d_in[18];

  float* ws   = (float*)d_ws;
  float* h    = ws + WS_H;
  float* xc   = ws + WS_XC;
  float* adj  = ws + WS_ADJ;
  float* invr = ws + WS_INVR;
  float* cs   = ws + WS_CS;
  float* haB  = ws + WS_HAB;
  float* S0   = ws + WS_S0;
  float* S1   = ws + WS_S1;
  float* agg  = ws + WS_AGG;

  float* out     = (float*)d_out;
  float* predOut = out;                                   // N x 2
  float* zOut    = out + (size_t)N_NODES * 2;             // N x 128
  float* avOut   = zOut + (size_t)N_NODES * NHID;         // 32 x 128
  float* alOut   = avOut + (size_t)NA * NHID;             // 32 x 2

  const int blocks2 = (NT + 1) / 2;  // 2 waves / 64-thread block, 1 tile/wave

  k_prep<<<1, 128, 0, stream>>>(anchors, ml_w1, ml_b1, cls_w1, cls_b1, prelu_a,
                                cls_w2, cls_b2, haB, cs, S0, S1, avOut, alOut);
  k_embed<<<blocks2, 64, 0, stream>>>(x, im_w1, im_b1, im_w2, im_b2, haB, ml_w1,
                                      ml_w2, ml_b2, h, adj, invr, cs);
  // hop 0
  k_hop_gather<<<256, 64, 0, stream>>>(h, adj, enc_w, S0);
  k_hop_mid<<<(NA * NHID + 127) / 128, 128, 0, stream>>>(S0, cs, agg);
  k_hop_apply<false><<<blocks2, 64, 0, stream>>>(
      adj, invr, agg, enc_b, h, ln_g, ln_b, cls_w1, cls_b1, prelu_a, cls_w2,
      cls_b2, xc, predOut, zOut);
  // hop 1 (final: residual + LN + classifier)
  k_hop_gather<<<256, 64, 0, stream>>>(xc, adj, enc_w + NHID * NHID, S1);
  k_hop_mid<<<(NA * NHID + 127) / 128, 128, 0, stream>>>(S1, cs, agg);
  k_hop_apply<true><<<blocks2, 64, 0, stream>>>(
      adj, invr, agg, enc_b + NHID, h, ln_g, ln_b, cls_w1, cls_b1, prelu_a,
      cls_w2, cls_b2, xc, predOut, zOut);
}